// NonLocalBlock_68470368633012
// MI455X (gfx1250) — compile-verified
//
#include <hip/hip_runtime.h>
#include <hip/hip_bf16.h>

// NonLocalBlock, fp32 end-to-end on CDNA5 WMMA (V_WMMA_F32_16X16X4_F32).
// b=8, c=64, ic=32, h=w=256, HW=65536, n=32768. Memory-bound (~0.7GB traffic,
// ~13 GFLOP) -> keep fp32, use WMMA for all matmul stages, LDS staging for
// operand reuse, fold softmax normalization + w_mask into tiny 64x64 matrices.

typedef float v2f __attribute__((ext_vector_type(2)));
typedef float v8f __attribute__((ext_vector_type(8)));

#define BATCH 8
#define CHN   64
#define ICHN  32
#define HW    65536
#define NHALF 32768

static __device__ __forceinline__ v8f wmma4(v2f a, v2f b, v8f c) {
  // D = A(16x4 f32) * B(4x16 f32) + C(16x16 f32)
  return __builtin_amdgcn_wmma_f32_16x16x4_f32(
      /*neg_a=*/false, a, /*neg_b=*/false, b,
      /*c_mod=*/(short)0, c, /*reuse_a=*/false, /*reuse_b=*/false);
}

// Monotonic uint key for float atomicMax (all finite floats map to >0 keys,
// so zero-initialized stats act as -inf).
static __device__ __forceinline__ unsigned fkey(float f) {
  unsigned u = __float_as_uint(f);
  return (u & 0x80000000u) ? ~u : (u | 0x80000000u);
}
static __device__ __forceinline__ float fdec(unsigned u) {
  return __uint_as_float((u & 0x80000000u) ? (u ^ 0x80000000u) : ~u);
}

// ---------------------------------------------------------------------------
// K1: three 1x1 convs (32x64)@(64x65536) per batch via WMMA; also per-batch
// running max of phi and g (atomicMax with uint key).
// grid (512, 8), block 256 (8 waves); each wave owns 16 pixels.
// ---------------------------------------------------------------------------
__global__ __launch_bounds__(256) void conv3_kernel(
    const float* __restrict__ x, const float* __restrict__ w_phi,
    const float* __restrict__ w_theta, const float* __restrict__ w_g,
    float* __restrict__ phi, float* __restrict__ g, float* __restrict__ theta,
    unsigned* __restrict__ stats) {
  __shared__ float sw[3 * ICHN * CHN];  // [phi | g | theta], 24 KB
  const int tid = threadIdx.x;
  const int b = blockIdx.y;
  for (int i = tid; i < ICHN * CHN; i += 256) {
    sw[i]        = w_phi[i];
    sw[2048 + i] = w_g[i];
    sw[4096 + i] = w_theta[i];
  }
  __syncthreads();

  const int lane = tid & 31, wave = tid >> 5;
  const long xb = (long)b * CHN * HW;
  const int n0 = blockIdx.x * 128 + wave * 16;
  const int bcol = n0 + (lane & 15);
  const int koff = (lane >> 4) * 2;  // lanes 0-15: K pair {0,1}; 16-31: {2,3}
  const int arow = lane & 15;

  v8f acc[6];  // [phi m0, phi m1, g m0, g m1, th m0, th m1]
  v8f z = {};
#pragma unroll
  for (int i = 0; i < 6; i++) acc[i] = z;

  for (int k0 = 0; k0 < CHN; k0 += 4) {
    v2f bf;  // B frag: rows = input channel, cols = pixel (coalesced 16 floats)
    bf.x = x[xb + (long)(k0 + koff + 0) * HW + bcol];
    bf.y = x[xb + (long)(k0 + koff + 1) * HW + bcol];
#pragma unroll
    for (int s = 0; s < 3; s++) {
      const float* wb = &sw[s * 2048];
#pragma unroll
      for (int m = 0; m < 2; m++) {
        const int oc = m * 16 + arow;
        v2f af;
        af.x = wb[oc * CHN + k0 + koff + 0];
        af.y = wb[oc * CHN + k0 + koff + 1];
        acc[s * 2 + m] = wmma4(af, bf, acc[s * 2 + m]);
      }
    }
  }

  // store: D layout VGPR r, lane l -> M = m*16 + r + 8*(l>>4), N = l&15
  float* outp[3] = {phi, g, theta};
  const long ob = (long)b * ICHN * HW;
#pragma unroll
  for (int s = 0; s < 3; s++)
#pragma unroll
    for (int m = 0; m < 2; m++)
#pragma unroll
      for (int r = 0; r < 8; r++) {
        const int oc = m * 16 + r + 8 * (lane >> 4);
        outp[s][ob + (long)oc * HW + bcol] = acc[s * 2 + m][r];
      }

  // per-thread max of phi / g tiles -> block reduce -> atomicMax
  float mp = -3.402823466e38f, mg = -3.402823466e38f;
#pragma unroll
  for (int r = 0; r < 8; r++) {
    mp = fmaxf(mp, fmaxf(acc[0][r], acc[1][r]));
    mg = fmaxf(mg, fmaxf(acc[2][r], acc[3][r]));
  }
  __syncthreads();  // done reading sw; reuse as reduction scratch
  sw[tid] = mp;
  sw[256 + tid] = mg;
  __syncthreads();
  for (int off = 128; off > 0; off >>= 1) {
    if (tid < off) {
      sw[tid] = fmaxf(sw[tid], sw[tid + off]);
      sw[256 + tid] = fmaxf(sw[256 + tid], sw[256 + tid + off]);
    }
    __syncthreads();
  }
  if (tid == 0) {
    atomicMax(&stats[b * 4 + 0], fkey(sw[0]));    // max phi
    atomicMax(&stats[b * 4 + 1], fkey(sw[256]));  // max g
  }
}

// ---------------------------------------------------------------------------
// K2: per-batch sum of exp(v - max) over phi (z=0) and g (z=1).
// grid (256, 8, 2), block 256.
// ---------------------------------------------------------------------------
__global__ __launch_bounds__(256) void sumexp_kernel(
    const float* __restrict__ phi, const float* __restrict__ g,
    unsigned* __restrict__ stats) {
  __shared__ float red[256];
  const int tid = threadIdx.x;
  const int b = blockIdx.y;
  const int which = blockIdx.z;
  const float* src = which ? g : phi;
  const float mx = fdec(stats[b * 4 + which]);
  const long base = (long)b * ICHN * HW;
  float s = 0.f;
  for (long i = (long)blockIdx.x * 256 + tid; i < (long)ICHN * HW;
       i += (long)gridDim.x * 256)
    s += __expf(src[base + i] - mx);
  red[tid] = s;
  __syncthreads();
  for (int off = 128; off > 0; off >>= 1) {
    if (tid < off) red[tid] += red[tid + off];
    __syncthreads();
  }
  if (tid == 0) atomicAdd((float*)&stats[b * 4 + 2 + which], red[0]);
}

// ---------------------------------------------------------------------------
// K3: att_raw[c,d] = sum_n exp(g1[c,n]-mg) * exp(phi1[d,n]-mp) on the
// reshaped (64, 32768) views. WMMA with K over n; exp applied while staging
// 64x128 chunks of each operand into LDS (64 KB). Partial tiles atomicAdd'ed.
// grid (8 k-chunks, 8 batches), block 256 (8 waves, 2 tiles each of 16).
// ---------------------------------------------------------------------------
__global__ __launch_bounds__(256) void att_kernel(
    const float* __restrict__ phi, const float* __restrict__ g,
    const unsigned* __restrict__ stats, float* __restrict__ att) {
  __shared__ float gex[64 * 128];
  __shared__ float pex[64 * 128];
  const int tid = threadIdx.x;
  const int b = blockIdx.y;
  const float mp = fdec(stats[b * 4 + 0]);
  const float mg = fdec(stats[b * 4 + 1]);
  const long base = (long)b * ICHN * HW;
  const int lane = tid & 31, wave = tid >> 5;
  const int t0 = wave * 2;  // 16 tiles (4x4 of 16x16) over 8 waves
  const int kspan = NHALF / gridDim.x;  // 4096
  const int kbase = blockIdx.x * kspan;
  const int koff = (lane >> 4) * 2;

  v8f acc[2];
  v8f z = {};
  acc[0] = z;
  acc[1] = z;

  for (int ch = 0; ch < kspan; ch += 128) {
    __syncthreads();
    for (int i = tid; i < 64 * 128; i += 256) {
      const int c = i >> 7, n = i & 127;
      // reshaped row c -> raw channel c>>1, pixel half c&1
      const long idx =
          base + (long)(c >> 1) * HW + (long)(c & 1) * NHALF + kbase + ch + n;
      gex[i] = __expf(g[idx] - mg);
      pex[i] = __expf(phi[idx] - mp);
    }
    __syncthreads();
#pragma unroll
    for (int tt = 0; tt < 2; tt++) {
      const int t = t0 + tt;
      const int mt = t >> 2, nt = t & 3;
      const int crow = mt * 16 + (lane & 15);
      const int drow = nt * 16 + (lane & 15);
      v8f a = acc[tt];
      for (int k0 = 0; k0 < 128; k0 += 4) {
        v2f af, bf;
        af.x = gex[crow * 128 + k0 + koff + 0];
        af.y = gex[crow * 128 + k0 + koff + 1];
        bf.x = pex[drow * 128 + k0 + koff + 0];
        bf.y = pex[drow * 128 + k0 + koff + 1];
        a = wmma4(af, bf, a);
      }
      acc[tt] = a;
    }
  }

#pragma unroll
  for (int tt = 0; tt < 2; tt++) {
    const int t = t0 + tt;
    const int mt = t >> 2, nt = t & 3;
#pragma unroll
    for (int r = 0; r < 8; r++) {
      const int c = mt * 16 + r + 8 * (lane >> 4);
      const int d = nt * 16 + (lane & 15);
      atomicAdd(&att[(long)b * 4096 + c * 64 + d], acc[tt][r]);
    }
  }
}

// ---------------------------------------------------------------------------
// K4: fold softmax scale + w_mask + reshape into two 64x64 matrices per batch:
// A[q][o][d] = (1/(S_phi*S_g)) * sum_i w_mask[o,i] * att[2i+q, d]
// grid (8), block 256. Tiny.
// ---------------------------------------------------------------------------
__global__ __launch_bounds__(256) void combine_kernel(
    const float* __restrict__ att, const unsigned* __restrict__ stats,
    const float* __restrict__ w_mask, float* __restrict__ A) {
  __shared__ float sa[4096];
  const int b = blockIdx.x, tid = threadIdx.x;
  const float sp = ((const float*)stats)[b * 4 + 2];
  const float sg = ((const float*)stats)[b * 4 + 3];
  const float s = 1.0f / (sp * sg);
  for (int i = tid; i < 4096; i += 256) sa[i] = att[(long)b * 4096 + i] * s;
  __syncthreads();
  for (int e = tid; e < 8192; e += 256) {
    const int q = e >> 12, o = (e >> 6) & 63, d = e & 63;
    float acc = 0.f;
    for (int i = 0; i < ICHN; i++)
      acc += w_mask[o * ICHN + i] * sa[((i << 1) | q) * 64 + d];
    A[(long)b * 8192 + e] = acc;
  }
}

// ---------------------------------------------------------------------------
// K5: out[o, q*32768+n] = sum_d A[q][o][d] * theta_reshaped[d, n]
// (2x 64x64)@(64x32768) per batch via WMMA, A staged in LDS (32 KB),
// coalesced fp32 stores of the final output.
// grid (256, 8), block 256 (8 waves x 16 cols).
// ---------------------------------------------------------------------------
__global__ __launch_bounds__(256) void final_kernel(
    const float* __restrict__ theta, const float* __restrict__ A,
    float* __restrict__ out) {
  __shared__ float sA[8192];  // [q][o][d]
  const int tid = threadIdx.x;
  const int b = blockIdx.y;
  for (int i = tid; i < 8192; i += 256) sA[i] = A[(long)b * 8192 + i];
  __syncthreads();

  const int lane = tid & 31, wave = tid >> 5;
  const int n0 = blockIdx.x * 128 + wave * 16;
  const int col = n0 + (lane & 15);
  const long tb = (long)b * ICHN * HW;
  const int koff = (lane >> 4) * 2;

  v8f acc[8];  // [q][m]
  v8f z = {};
#pragma unroll
  for (int i = 0; i < 8; i++) acc[i] = z;

  for (int k0 = 0; k0 < 64; k0 += 4) {
    const int d0 = k0 + koff, d1 = d0 + 1;
    v2f bf;  // theta_reshaped[d, col]: raw channel d>>1, pixel half d&1
    bf.x = theta[tb + (long)(d0 >> 1) * HW + (long)(d0 & 1) * NHALF + col];
    bf.y = theta[tb + (long)(d1 >> 1) * HW + (long)(d1 & 1) * NHALF + col];
#pragma unroll
    for (int q = 0; q < 2; q++)
#pragma unroll
      for (int m = 0; m < 4; m++) {
        const int o = m * 16 + (lane & 15);
        v2f af;
        af.x = sA[q * 4096 + o * 64 + d0];
        af.y = sA[q * 4096 + o * 64 + d1];
        acc[q * 4 + m] = wmma4(af, bf, acc[q * 4 + m]);
      }
  }

  const long ob = (long)b * CHN * HW;
#pragma unroll
  for (int q = 0; q < 2; q++)
#pragma unroll
    for (int m = 0; m < 4; m++)
#pragma unroll
      for (int r = 0; r < 8; r++) {
        const int o = m * 16 + r + 8 * (lane >> 4);
        out[ob + (long)o * HW + (long)q * NHALF + col] = acc[q * 4 + m][r];
      }
}

// ---------------------------------------------------------------------------
extern "C" void kernel_launch(void* const* d_in, const int* in_sizes, int n_in,
                              void* d_out, int out_size, void* d_ws,
                              size_t ws_size, hipStream_t stream) {
  const float* x       = (const float*)d_in[0];
  const float* w_phi   = (const float*)d_in[1];
  const float* w_theta = (const float*)d_in[2];
  const float* w_g     = (const float*)d_in[3];
  const float* w_mask  = (const float*)d_in[4];
  float* out = (float*)d_out;

  // workspace layout: phi | g | theta (64 MB each) | stats | att | A
  const long CONV_ELEMS = (long)BATCH * ICHN * HW;  // 16,777,216 floats
  float* phi   = (float*)d_ws;
  float* g     = phi + CONV_ELEMS;
  float* theta = g + CONV_ELEMS;
  unsigned* stats = (unsigned*)(theta + CONV_ELEMS);  // 8 x 4 u32
  float* att = (float*)(stats + BATCH * 4);           // 8 x 64 x 64
  float* A   = att + (long)BATCH * 4096;              // 8 x 2 x 64 x 64

  // zero stats (max keys + sums) and att accumulator (contiguous region)
  hipMemsetAsync(stats, 0, BATCH * 4 * sizeof(unsigned) +
                               (size_t)BATCH * 4096 * sizeof(float), stream);

  conv3_kernel<<<dim3(512, BATCH), 256, 0, stream>>>(
      x, w_phi, w_theta, w_g, phi, g, theta, stats);
  sumexp_kernel<<<dim3(256, BATCH, 2), 256, 0, stream>>>(phi, g, stats);
  att_kernel<<<dim3(8, BATCH), 256, 0, stream>>>(phi, g, stats, att);
  combine_kernel<<<BATCH, 256, 0, stream>>>(att, stats, w_mask, A);
  final_kernel<<<dim3(256, BATCH), 256, 0, stream>>>(theta, A, out);
}